// GradientLayer_17729624998206
// MI455X (gfx1250) — compile-verified
//
#include <hip/hip_runtime.h>

// ---------------------------------------------------------------------------
// Edge-MLP (GNN edge network) for MI455X / gfx1250, wave32.
//
//   out[e,:] = a_e * (x_dst - b_e * x_src),  [a_e,b_e] = MLP(concat(x_dst,x_src))
//
// Strategy:
//  * 16 edges per wave  ->  first GEMM is (16x32)x(32x64) done with
//    V_WMMA_F32_16X16X4_F32 : 8 K-steps x 4 N-tiles = 32 WMMAs / tile.
//    fp32 WMMA keeps exact reference precision (inputs are fp32).
//  * x (6.4 MB) lives in the 192 MB L2, so the random gathers are L2 hits;
//    HBM traffic is dominated by the 205 MB output -> ~10us roofline floor.
//  * W1 fragments (8 KB) preloaded into VGPRs once, amortized over a
//    grid-stride loop of edge tiles.
//  * Second GEMM (64->2) is tiny: per-lane FMA + 4-step shfl_xor reduction
//    inside each 16-lane half (matches the WMMA C-layout M-partition).
// ---------------------------------------------------------------------------

typedef __attribute__((ext_vector_type(2))) float v2f;
typedef __attribute__((ext_vector_type(8))) float v8f;

#define FF 16   // node feature dim
#define HH 64   // hidden dim

__global__ __launch_bounds__(256) void edge_mlp_wmma_kernel(
    const float* __restrict__ x,      // (N,16) fp32
    const int*   __restrict__ eidx,   // (2,E) int32 : row0 = src, row1 = dst
    const float* __restrict__ W1,     // (32,64) fp32, row-major (k major)
    const float* __restrict__ b1,     // (64,)
    const float* __restrict__ W2,     // (64,2)
    const float* __restrict__ b2,     // (2,)
    float*       __restrict__ out,    // (E,16) fp32
    int E_total)
{
  const int lane = threadIdx.x & 31;
  const int wave = threadIdx.x >> 5;
  const int lm   = lane & 15;   // N (or M) position within a 16-lane half
  const int hi   = lane >> 4;   // half-wave select (K+2 offset in A/B frags)

  // ---- Preload W1 as WMMA B-fragments (K x N = 4 x 16 per step/tile). ----
  // B layout (mirrors documented A layout): lane L holds
  //   b.x = B[k0 + 2*hi][n],  b.y = B[k0 + 1 + 2*hi][n],  n = t*16 + lm
  v2f bfrag[8][4];
#pragma unroll
  for (int s = 0; s < 8; ++s) {
#pragma unroll
    for (int t = 0; t < 4; ++t) {
      const float* p = W1 + (4 * s + 2 * hi) * HH + t * 16 + lm;
      bfrag[s][t].x = p[0];
      bfrag[s][t].y = p[HH];
    }
  }
  // Bias / second-layer weights in the C-layout column (n = t*16 + lm).
  float b1v[4], w2a[4], w2b[4];
#pragma unroll
  for (int t = 0; t < 4; ++t) {
    const int n = t * 16 + lm;
    b1v[t] = b1[n];
    w2a[t] = W2[2 * n + 0];
    w2b[t] = W2[2 * n + 1];
  }
  const float b2a = b2[0];
  const float b2b = b2[1];

  const int ntiles = (E_total + 15) >> 4;
  const int gwave  = blockIdx.x * (blockDim.x >> 5) + wave;
  const int nwaves = gridDim.x * (blockDim.x >> 5);

  for (int tile = gwave; tile < ntiles; tile += nwaves) {
    const int e0 = tile << 4;
    int e = e0 + lm;
    if (e >= E_total) e = E_total - 1;           // clamp (E % 16 == 0 anyway)

    const int sidx = eidx[e];                    // src node
    const int didx = eidx[E_total + e];          // dst node
    const float* prow_d = x + (long)didx * FF;
    const float* prow_s = x + (long)sidx * FF;

    // ---- A-fragments: net_in = [x_dst | x_src], step s covers K=4s..4s+3.
    // Lane L supplies net_in[m][4s+2hi], net_in[m][4s+2hi+1]  (m = lm):
    // one 8-byte load per step; the two halves cover the row exactly once.
    v2f afrag[8];
#pragma unroll
    for (int s = 0; s < 4; ++s)
      afrag[s] = *(const v2f*)(prow_d + 4 * s + 2 * hi);
#pragma unroll
    for (int s = 4; s < 8; ++s)
      afrag[s] = *(const v2f*)(prow_s + 4 * (s - 4) + 2 * hi);

    // ---- First GEMM: 32 x V_WMMA_F32_16X16X4_F32 (exact fp32). ----
    v8f acc[4] = {};
#pragma unroll
    for (int s = 0; s < 8; ++s) {
#pragma unroll
      for (int t = 0; t < 4; ++t) {
        acc[t] = __builtin_amdgcn_wmma_f32_16x16x4_f32(
            /*neg_a=*/false, afrag[s],
            /*neg_b=*/false, bfrag[s][t],
            /*c_mod=*/(short)0, acc[t],
            /*reuse_a=*/false, /*reuse_b=*/false);
      }
    }

    // ---- Bias + ReLU + second GEMM (64 -> 2), fused in C-layout. ----
    // C-layout: acc[t][r] = h_pre[m = r + 8*hi][n = t*16 + lm].
    float pa[8], pb[8];
#pragma unroll
    for (int r = 0; r < 8; ++r) { pa[r] = 0.0f; pb[r] = 0.0f; }
#pragma unroll
    for (int t = 0; t < 4; ++t) {
#pragma unroll
      for (int r = 0; r < 8; ++r) {
        float h = acc[t][r] + b1v[t];
        h = h > 0.0f ? h : 0.0f;                 // ReLU
        pa[r] = __builtin_fmaf(h, w2a[t], pa[r]);
        pb[r] = __builtin_fmaf(h, w2b[t], pb[r]);
      }
    }
    // Reduce over n across the 16 lanes of each half (xor masks < 16 keep
    // the half-wave, i.e. the M-partition, intact).
#pragma unroll
    for (int mask = 1; mask < 16; mask <<= 1) {
#pragma unroll
      for (int r = 0; r < 8; ++r) {
        pa[r] += __shfl_xor(pa[r], mask, 32);
        pb[r] += __shfl_xor(pb[r], mask, 32);
      }
    }

    // ---- Elementwise output: lane L writes feature f = lm of edge
    // m = r + 8*hi (exactly the edge its reduced a/b belong to).
    // 16 lanes -> 64B coalesced stores per edge row.
#pragma unroll
    for (int r = 0; r < 8; ++r) {
      const int   mm = hi * 8 + r;
      const float am = pa[r] + b2a;
      const float bm = pb[r] + b2b;
      const int d2 = __shfl(didx, mm, 32);       // lane mm holds edge e0+mm
      const int s2 = __shfl(sidx, mm, 32);
      const int eo = e0 + mm;
      if (eo < E_total) {
        const float xd = x[(long)d2 * FF + lm];
        const float xs = x[(long)s2 * FF + lm];
        out[(long)eo * FF + lm] = am * (xd - bm * xs);
      }
    }
  }
}

extern "C" void kernel_launch(void* const* d_in, const int* in_sizes, int n_in,
                              void* d_out, int out_size, void* d_ws, size_t ws_size,
                              hipStream_t stream) {
  const float* x    = (const float*)d_in[0];
  const int*   eidx = (const int*)d_in[1];   // JAX default x64-off => int32
  const float* W1   = (const float*)d_in[2];
  const float* b1   = (const float*)d_in[3];
  const float* W2   = (const float*)d_in[4];
  const float* b2   = (const float*)d_in[5];
  float*       out  = (float*)d_out;

  const int E_total = in_sizes[1] / 2;       // edge_index is (2, E)

  // 8 waves / block; grid-stride over E/16 = 200k edge tiles so the
  // per-wave W1 register preload is amortized (~8 tiles per wave at 3072 CTAs).
  dim3 block(256);
  dim3 grid(3072);
  hipLaunchKernelGGL(edge_mlp_wmma_kernel, grid, block, 0, stream,
                     x, eidx, W1, b1, W2, b2, out, E_total);
}